// Crystal_Norm_46248207843552
// MI455X (gfx1250) — compile-verified
//
#include <hip/hip_runtime.h>
#include <stdint.h>

#define SEGS   16384
#define FEA    128
#define MAXR   120          // cached rows per segment: 120*128*4 = 60 KB LDS
#define EPS_   1e-6f
#define VFLOOR 1e-7f

typedef __attribute__((ext_vector_type(2))) float        v2f;
typedef __attribute__((ext_vector_type(4))) float        f4v;
typedef __attribute__((ext_vector_type(8))) float        v8f;
typedef __attribute__((ext_vector_type(4))) unsigned int v4u;
typedef __attribute__((ext_vector_type(8))) int          v8i;
typedef __attribute__((ext_vector_type(4))) int          v4i;

#if __has_builtin(__builtin_amdgcn_tensor_load_to_lds) && __has_builtin(__builtin_amdgcn_s_wait_tensorcnt)
#define HAVE_TDM 1
#else
#define HAVE_TDM 0
#endif

#if __has_builtin(__builtin_amdgcn_wmma_f32_16x16x4_f32)
#define HAVE_WMMA 1
#else
#define HAVE_WMMA 0
#endif

// -------- kernel 1: segment bounds via binary search over sorted index ------
__global__ void seg_bounds_kernel(const int* __restrict__ index,
                                  int* __restrict__ seg_start, int n) {
    int s = blockIdx.x * blockDim.x + threadIdx.x;
    if (s > SEGS) return;
    if (s == SEGS) { seg_start[SEGS] = n; return; }
    int lo = 0, hi = n;
    while (lo < hi) {
        int mid = (lo + hi) >> 1;
        if (index[mid] < s) lo = mid + 1; else hi = mid;
    }
    seg_start[s] = lo;
}

// -------- kernel 2: one block per segment --------------------------------
__global__ __launch_bounds__(256) void crystal_norm_kernel(
    const float* __restrict__ x, const int* __restrict__ seg_start,
    const float* __restrict__ weight, const float* __restrict__ bias,
    float* __restrict__ out) {

    __shared__ alignas(16) float tile[MAXR * FEA];   // 60 KB
    __shared__ alignas(16) float s_mean[FEA];
    __shared__ alignas(16) float s_scale[FEA];

    const int s     = blockIdx.x;
    const int start = seg_start[s];
    const int end   = seg_start[s + 1];
    const int n     = end - start;
    if (n <= 0) return;                  // uniform per block
    const int cached = (n < MAXR) ? n : MAXR;
    const int t      = threadIdx.x;
    const float* xb  = x + (size_t)start * FEA;
    float* ob        = out + (size_t)start * FEA;

    // ---- stage rows into LDS via the Tensor Data Mover (one HBM read) ----
#if HAVE_TDM
    if (t < 32) {                         // wave 0 issues the TDM descriptor
        unsigned long long ga = (unsigned long long)(uintptr_t)xb;
        unsigned int lds_off  = (unsigned int)(unsigned long long)(uintptr_t)&tile[0];
        v4u g0;
        g0[0] = 1u;                                            // count=1, user desc
        g0[1] = lds_off;                                       // lds_addr
        g0[2] = (unsigned int)(ga & 0xFFFFFFFFu);              // global_addr lo
        g0[3] = (unsigned int)((ga >> 32) & 0x1FFFFFFu) | (2u << 30); // hi + type=2
        v8i g1 = {0,0,0,0,0,0,0,0};
        g1[0] = (int)(2u << 16);                               // data_size = 4B
        g1[1] = (int)((unsigned)FEA << 16);                    // tensor_dim0 = 128
        g1[2] = (int)(((unsigned)cached & 0xFFFFu) << 16);     // tensor_dim1 = n
        g1[3] = (int)((unsigned)FEA << 16);                    // tile_dim0 = 128
        g1[4] = (int)((unsigned)cached & 0xFFFFu);             // tile_dim1 = n
        g1[5] = FEA;                                           // dim0_stride = 128
        v4i gz4 = {0,0,0,0};
        v8i gz8 = {0,0,0,0,0,0,0,0};
        __builtin_amdgcn_tensor_load_to_lds(g0, g1, gz4, gz4, gz8, 0);
        __builtin_amdgcn_s_wait_tensorcnt(0);
    }
#else
    for (int c = t; c < cached * (FEA / 4); c += 256) {
        f4v v = ((const f4v*)xb)[c];
        ((f4v*)tile)[c] = v;
    }
#endif
    __syncthreads();

    if (n <= MAXR) {
        // ================= FAST PATH: whole segment resident in LDS =========
#if HAVE_WMMA
        // WMMA f32 16x16x4: A = ones(16x4); D row 0 = per-feature column sums.
        const int w    = t >> 5;           // wave 0..7 -> features [16w, 16w+16)
        const int lane = t & 31;
        const int f0   = w * 16;
        const int fl   = f0 + (lane & 15);
        const int hb   = (lane >> 4) * 2;  // B: VGPR0={K0|K2}, VGPR1={K1|K3}
        const int nal  = n & ~3;
        v2f ones; ones[0] = 1.0f; ones[1] = 1.0f;

        v8f acc = {0,0,0,0,0,0,0,0};
        const float* bp = &tile[hb * FEA + fl];
        for (int r = 0; r < nal; r += 4) {
            v2f b;
            b[0] = bp[(r    ) * FEA];
            b[1] = bp[(r + 1) * FEA];
            acc = __builtin_amdgcn_wmma_f32_16x16x4_f32(
                false, ones, false, b, (short)0, acc, false, false);
        }
        if (nal < n) {                    // 1-3 tail rows, zero-padded
            int r0 = nal + hb;
            v2f b;
            b[0] = (r0     < n) ? tile[r0 * FEA + fl]       : 0.0f;
            b[1] = (r0 + 1 < n) ? tile[(r0 + 1) * FEA + fl] : 0.0f;
            acc = __builtin_amdgcn_wmma_f32_16x16x4_f32(
                false, ones, false, b, (short)0, acc, false, false);
        }
        if (lane < 16) s_mean[f0 + lane] = acc[0] / (float)n;  // D[M=0][N=lane]
        __syncthreads();

        v8f acc2 = {0,0,0,0,0,0,0,0};
        const float mloc = s_mean[fl];
        for (int r = 0; r < nal; r += 4) {
            float d0 = bp[(r    ) * FEA] - mloc;
            float d1 = bp[(r + 1) * FEA] - mloc;
            v2f b; b[0] = d0 * d0; b[1] = d1 * d1;
            acc2 = __builtin_amdgcn_wmma_f32_16x16x4_f32(
                false, ones, false, b, (short)0, acc2, false, false);
        }
        if (nal < n) {
            int r0 = nal + hb;
            float d0 = (r0     < n) ? (tile[r0 * FEA + fl]       - mloc) : 0.0f;
            float d1 = (r0 + 1 < n) ? (tile[(r0 + 1) * FEA + fl] - mloc) : 0.0f;
            v2f b; b[0] = d0 * d0; b[1] = d1 * d1;
            acc2 = __builtin_amdgcn_wmma_f32_16x16x4_f32(
                false, ones, false, b, (short)0, acc2, false, false);
        }
        if (lane < 16) {
            int f = f0 + lane;
            float var = (acc2[0] + EPS_) / (float)(n - 1);  // n==1 -> inf -> scale 0
            float sd  = sqrtf(fmaxf(var, VFLOOR));
            s_scale[f] = weight[f] / (sd + EPS_);
        }
        __syncthreads();
#else
        if (t < FEA) {
            float sum = 0.0f;
            for (int r = 0; r < n; ++r) sum += tile[r * FEA + t];
            s_mean[t] = sum / (float)n;
        }
        __syncthreads();
        if (t < FEA) {
            float m = s_mean[t], ds = 0.0f;
            for (int r = 0; r < n; ++r) { float d = tile[r * FEA + t] - m; ds = fmaf(d, d, ds); }
            float var = (ds + EPS_) / (float)(n - 1);
            s_scale[t] = weight[t] / (sqrtf(fmaxf(var, VFLOOR)) + EPS_);
        }
        __syncthreads();
#endif
        // normalize: stride-256 over float4s => each thread owns 4 fixed features
        const int g  = t & 31;                 // float4 group within a row
        const f4v m4 = ((const f4v*)s_mean)[g];
        const f4v s4 = ((const f4v*)s_scale)[g];
        const f4v b4 = ((const f4v*)bias)[g];
        const f4v* t4 = (const f4v*)tile;
        f4v* o4       = (f4v*)ob;
        const int tot4 = n * (FEA / 4);
        for (int i4 = t; i4 < tot4; i4 += 256) {
            f4v v = t4[i4];
            f4v o = (v - m4) * s4 + b4;
            __builtin_nontemporal_store(o, &o4[i4]);
        }
    } else {
        // ================= SLOW PATH: segment overflows LDS (P ~ 1e-14) =====
        auto loadx = [&](int r, int f) -> float {
            return (r < cached) ? tile[r * FEA + f] : xb[(size_t)r * FEA + f];
        };
        if (t < FEA) {
            float sum = 0.0f;
            for (int r = 0; r < n; ++r) sum += loadx(r, t);
            s_mean[t] = sum / (float)n;
        }
        __syncthreads();
        if (t < FEA) {
            float m = s_mean[t], ds = 0.0f;
            for (int r = 0; r < n; ++r) { float d = loadx(r, t) - m; ds = fmaf(d, d, ds); }
            float var = (ds + EPS_) / (float)(n - 1);
            s_scale[t] = weight[t] / (sqrtf(fmaxf(var, VFLOOR)) + EPS_);
        }
        __syncthreads();
        const int total = n * FEA;
        for (int i = t; i < total; i += 256) {
            int r = i >> 7, f = i & (FEA - 1);
            float xv = loadx(r, f);
            float o  = (xv - s_mean[f]) * s_scale[f] + bias[f];
            __builtin_nontemporal_store(o, &ob[i]);
        }
    }
}

extern "C" void kernel_launch(void* const* d_in, const int* in_sizes, int n_in,
                              void* d_out, int out_size, void* d_ws, size_t ws_size,
                              hipStream_t stream) {
    (void)n_in; (void)out_size; (void)ws_size;
    const float* target_fea = (const float*)d_in[0];
    const int*   index      = (const int*)d_in[1];
    const float* weight     = (const float*)d_in[2];
    const float* bias       = (const float*)d_in[3];
    float*       out        = (float*)d_out;
    int*         seg_start  = (int*)d_ws;          // (SEGS+1) ints
    const int N = in_sizes[1];

    seg_bounds_kernel<<<(SEGS + 1 + 255) / 256, 256, 0, stream>>>(index, seg_start, N);
    crystal_norm_kernel<<<SEGS, 256, 0, stream>>>(target_fea, seg_start, weight, bias, out);
}